// UAG_RNN_4Neigh_15908558864728
// MI455X (gfx1250) — compile-verified
//
#include <hip/hip_runtime.h>

typedef float v2f __attribute__((ext_vector_type(2)));
typedef float v8f __attribute__((ext_vector_type(8)));

#define CH   64
#define HH   160
#define WW   160
#define SPP  (HH * WW)          // per-channel plane size
#define NPIX (8 * CH * SPP)     // total elements of x / hs / hn / out

struct DirParams {
    const float* Wc;   // weight applied to current input
    const float* Wp;   // weight applied to previous state
    const float* ba;   // bias a
    const float* bb;   // bias b
    const float* src;  // input tensor for this scan
    float*       dst;  // output tensor (write or accumulate)
    int reverse;       // 1 = scan high->low position
    int relu_init;     // 1 = relu() the initial slice
    int accumulate;    // 1 = dst += val, 0 = dst = val
    int step_stride;   // element stride along scan direction (rows: W, cols: 1)
    int n_stride;      // element stride along the 16-seq dim (rows: 1, cols: W)
};

// One wave owns 16 independent sequences. Both 64x64 weight matrices are held
// in registers as WMMA A-fragments (256 VGPRs; gfx1250 waves address up to 1024
// VGPRs via the VGPR-MSB mechanism). Recurrent state [64ch x 16seq] ping-pongs
// between two per-wave LDS buffers. Per step: 128 v_wmma_f32_16x16x4_f32 in
// 8 independent 16-deep accumulation chains (Wc-chain fed by global loads,
// Wp-chain fed by LDS loads).
__global__ void __launch_bounds__(128, 1)
uag_scan_kernel(DirParams pa, DirParams pb) {
    const DirParams P = (blockIdx.y == 0) ? pa : pb;

    __shared__ float sX[4][2][CH * 16];   // per-wave ping-pong recurrent state

    const int tid  = threadIdx.x;
    const int wave = tid >> 5;            // wave32
    const int lane = tid & 31;
    const int g    = lane >> 4;           // lane half (selects K pair / M half)
    const int n    = lane & 15;           // B/D column (sequence within tile)
    const int ml   = lane & 15;           // A row within tile

    const int seq0 = (blockIdx.x * 4 + wave) * 16;
    const int b  = seq0 / 160;
    const int p0 = seq0 - b * 160;        // 160 % 16 == 0 -> 16 seqs share b

    // ---- hoist both weight matrices into A-fragment registers (loop-invariant) ----
    v2f wc[4][16], wp[4][16];
#pragma unroll
    for (int t = 0; t < 4; ++t)
#pragma unroll
        for (int kf = 0; kf < 16; ++kf) {
            const int m  = t * 16 + ml;
            const int c0 = kf * 4 + g * 2;
            wc[t][kf].x = P.Wc[m * 64 + c0];
            wc[t][kf].y = P.Wc[m * 64 + c0 + 1];
            wp[t][kf].x = P.Wp[m * 64 + c0];
            wp[t][kf].y = P.Wp[m * 64 + c0 + 1];
        }

    // Bias pre-broadcast in C/D fragment layout: element(m,n) -> vgpr r, m = 8*g + r.
    v8f bias[4];
#pragma unroll
    for (int t = 0; t < 4; ++t)
#pragma unroll
        for (int r = 0; r < 8; ++r) {
            int c = t * 16 + g * 8 + r;
            bias[t][r] = P.ba[c] + P.bb[c];
        }

    float* s0 = &sX[wave][0][0];
    float* s1 = &sX[wave][1][0];

    // ---- initial slice: load, optional relu, write out, seed state buffer 0 ----
    const int pos0 = P.reverse ? 159 : 0;
    for (int i2 = 0; i2 < 32; ++i2) {
        int idx = i2 * 32 + lane;                 // 0..1023 over [c][n]
        int c = idx >> 4, nn = idx & 15;
        int a = b * CH * SPP + c * SPP + pos0 * P.step_stride + (p0 + nn) * P.n_stride;
        float v = P.src[a];
        if (P.relu_init) v = fmaxf(v, 0.0f);
        s0[idx] = v;
        if (P.accumulate) P.dst[a] += v; else P.dst[a] = v;
    }

    const int cb = b * CH * SPP + (p0 + n) * P.n_stride;

    // ---- recurrent scan: 159 dependent steps ----
    for (int i = 1; i < 160; ++i) {
        const int pos   = P.reverse ? (159 - i) : i;
        const int sbase = cb + pos * P.step_stride;
        float* sPrev = (i & 1) ? s0 : s1;         // step i reads buf[(i-1)&1]
        float* sNext = (i & 1) ? s1 : s0;

        v8f accC[4] = { bias[0], bias[1], bias[2], bias[3] };  // Wc@cur chain (+bias)
        v8f accP[4] = {};                                      // Wp@prev chain

#pragma unroll
        for (int kf = 0; kf < 16; ++kf) {
            const int c0 = kf * 4 + g * 2;        // this lane's K pair within the chunk
            v2f bc, bp;
            bc.x = P.src[sbase + (c0 + 0) * SPP];
            bc.y = P.src[sbase + (c0 + 1) * SPP];
            bp.x = sPrev[(c0 + 0) * 16 + n];
            bp.y = sPrev[(c0 + 1) * 16 + n];
#pragma unroll
            for (int t = 0; t < 4; ++t) {         // 8 independent 16-deep chains
                accC[t] = __builtin_amdgcn_wmma_f32_16x16x4_f32(
                    false, wc[t][kf], false, bc, (short)0, accC[t], false, false);
                accP[t] = __builtin_amdgcn_wmma_f32_16x16x4_f32(
                    false, wp[t][kf], false, bp, (short)0, accP[t], false, false);
            }
        }

        // Combine chains, ReLU, write out, recycle into the other state buffer.
        const int abase = b * CH * SPP + pos * P.step_stride + (p0 + n) * P.n_stride;
#pragma unroll
        for (int t = 0; t < 4; ++t)
#pragma unroll
            for (int r = 0; r < 8; ++r) {
                float v = fmaxf(accC[t][r] + accP[t][r], 0.0f);
                int c = t * 16 + g * 8 + r;
                int a = abase + c * SPP;
                if (P.accumulate) P.dst[a] += v; else P.dst[a] = v;
                sNext[c * 16 + n] = v;
            }

        // Prefetch next step's activations into cache (global_prefetch_b8).
        if (i < 159) {
            int npos = P.reverse ? (159 - (i + 1)) : (i + 1);
            __builtin_prefetch(&P.src[cb + npos * P.step_stride], 0, 1);
        }
    }
}

extern "C" void kernel_launch(void* const* d_in, const int* in_sizes, int n_in,
                              void* d_out, int out_size, void* d_ws, size_t ws_size,
                              hipStream_t stream) {
    (void)in_sizes; (void)n_in; (void)out_size; (void)ws_size;

    const float* x   = (const float*)d_in[0];
    const float* w1  = (const float*)d_in[1];  const float* b1  = (const float*)d_in[2];
    const float* w2  = (const float*)d_in[3];  const float* b2  = (const float*)d_in[4];
    const float* w4  = (const float*)d_in[5];  const float* b4  = (const float*)d_in[6];
    const float* w5  = (const float*)d_in[7];  const float* b5  = (const float*)d_in[8];
    const float* w7  = (const float*)d_in[9];  const float* b7  = (const float*)d_in[10];
    const float* w8  = (const float*)d_in[11]; const float* b8  = (const float*)d_in[12];
    const float* w9  = (const float*)d_in[13]; const float* b9  = (const float*)d_in[14];
    const float* w10 = (const float*)d_in[15]; const float* b10 = (const float*)d_in[16];
    const float* w12 = (const float*)d_in[17]; const float* b12 = (const float*)d_in[18];
    const float* w13 = (const float*)d_in[19]; const float* b13 = (const float*)d_in[20];
    const float* w15 = (const float*)d_in[21]; const float* b15 = (const float*)d_in[22];
    const float* w16 = (const float*)d_in[23]; const float* b16 = (const float*)d_in[24];

    float* out = (float*)d_out;
    float* hs  = (float*)d_ws;           // south-scan result
    float* hn  = hs + (size_t)NPIX;      // north-scan result

    hipMemsetAsync(d_out, 0, (size_t)NPIX * sizeof(float), stream);

    // Row scans (south + north concurrently; 1280 seqs each = 20 blocks x 4 waves).
    DirParams south = { w1, w2,  b1,  b2,  x,  hs,  0, 0, 0, WW, 1 };
    DirParams north = { w9, w10, b9,  b10, x,  hn,  1, 1, 0, WW, 1 };
    uag_scan_kernel<<<dim3(20, 2), 128, 0, stream>>>(south, north);

    // Diagonal column scans accumulate into out (each owns disjoint elements per launch).
    DirParams se = { w4,  w5,  b4,  b5,  hs, out, 0, 1, 1, 1, WW };
    uag_scan_kernel<<<dim3(20, 1), 128, 0, stream>>>(se, se);
    DirParams sw = { w7,  w8,  b7,  b8,  hs, out, 1, 1, 1, 1, WW };
    uag_scan_kernel<<<dim3(20, 1), 128, 0, stream>>>(sw, sw);
    DirParams ne = { w12, w13, b12, b13, hn, out, 0, 1, 1, 1, WW };
    uag_scan_kernel<<<dim3(20, 1), 128, 0, stream>>>(ne, ne);
    DirParams nw = { w15, w16, b15, b16, hn, out, 1, 1, 1, 1, WW };
    uag_scan_kernel<<<dim3(20, 1), 128, 0, stream>>>(nw, nw);
}